// PointNet_pp_80942953661135
// MI455X (gfx1250) — compile-verified
//
#include <hip/hip_runtime.h>
#include <math.h>

#define INV_STD 0.9999950000374997f

typedef __attribute__((ext_vector_type(2))) float v2f;
typedef __attribute__((ext_vector_type(8))) float v8f;

__device__ inline v8f wmma4(v2f a, v2f b, v8f c) {
  // V_WMMA_F32_16X16X4_F32 : D = A(16x4) * B(4x16) + C(16x16), all f32
  return __builtin_amdgcn_wmma_f32_16x16x4_f32(false, a, false, b, (short)0, c, false, false);
}

// ---------------------------------------------------------------------------
// Wave-level fused MLP layer: O[rowsT*16, Cout] = relu((X @ W)*g*INV_STD + b)
// X in LDS (row-major, stride ldX, zero-padded to CinIter), W/g/b in global.
// A-frag layout (16x4 f32): lanes 0-15 -> K={k0,k0+1}, lanes 16-31 -> K={k0+2,k0+3}
// B-frag layout (4x16 f32): mirrors A; C frag: lane holds col n0+lane%16,
// rows r+(lane<16?0:8).
// ---------------------------------------------------------------------------
__device__ void mlp_layer_lds(const float* __restrict__ X, int ldX, int CinReal,
                              int CinIter, const float* __restrict__ W,
                              const float* __restrict__ G, const float* __restrict__ Bt,
                              float* __restrict__ O, int Cout, int rowsT) {
  const int lane = threadIdx.x & 31;
  const int wave = threadIdx.x >> 5;
  const int nw   = blockDim.x >> 5;
  const int ntC  = Cout >> 4;
  const int ntiles = rowsT * ntC;
  const int ksel = (lane >> 4) << 1;          // 0 or 2
  for (int t = wave; t < ntiles; t += nw) {   // wave-uniform loop (EXEC all ones)
    const int rt = t / ntC, ct = t - rt * ntC;
    const int mrow = (rt << 4) + (lane & 15);
    const int ncol = (ct << 4) + (lane & 15);
    v8f acc = {0.f, 0.f, 0.f, 0.f, 0.f, 0.f, 0.f, 0.f};
    for (int k0 = 0; k0 < CinIter; k0 += 4) {
      const int kb = k0 + ksel;
      v2f a, bf;
      a.x  = X[mrow * ldX + kb];
      a.y  = X[mrow * ldX + kb + 1];
      bf.x = (kb     < CinReal) ? W[(size_t)kb * Cout + ncol]       : 0.f;
      bf.y = (kb + 1 < CinReal) ? W[(size_t)(kb + 1) * Cout + ncol] : 0.f;
      acc = wmma4(a, bf, acc);
    }
    const float s  = G[ncol] * INV_STD;
    const float bb = Bt[ncol];
    const int rbase = (rt << 4) + ((lane >> 4) << 3);
#pragma unroll
    for (int r = 0; r < 8; ++r)
      O[(rbase + r) * Cout + ncol] = fmaxf(acc[r] * s + bb, 0.f);
  }
}

// ---------------------------------------------------------------------------
// Farthest point sampling: one block per batch, serial M steps.
// ---------------------------------------------------------------------------
__global__ void fps_kernel(const float* __restrict__ xyz, int N, int M,
                           int* __restrict__ idx, float* __restrict__ distws) {
  const int b = blockIdx.x;
  const float* X = xyz + (size_t)b * N * 3;
  float* dist = distws + (size_t)b * N;
  __shared__ float sval[256];
  __shared__ int   sidx[256];
  const int t = threadIdx.x, T = blockDim.x;
  for (int i = t; i < N; i += T) dist[i] = 1e10f;
  __syncthreads();
  int far = 0;
  for (int m = 0; m < M; ++m) {
    if (t == 0) idx[b * M + m] = far;
    const float cx = X[far * 3 + 0], cy = X[far * 3 + 1], cz = X[far * 3 + 2];
    float best = -1.f; int bi = 0;
    for (int i = t; i < N; i += T) {
      const float dx = X[i * 3 + 0] - cx, dy = X[i * 3 + 1] - cy, dz = X[i * 3 + 2] - cz;
      float dm = fminf(dist[i], dx * dx + dy * dy + dz * dz);
      dist[i] = dm;
      if (dm > best) { best = dm; bi = i; }
    }
    sval[t] = best; sidx[t] = bi;
    __syncthreads();
    for (int s = T >> 1; s > 0; s >>= 1) {
      if (t < s) {
        if (sval[t + s] > sval[t] ||
            (sval[t + s] == sval[t] && sidx[t + s] < sidx[t])) {
          sval[t] = sval[t + s]; sidx[t] = sidx[t + s];
        }
      }
      __syncthreads();
    }
    far = sidx[0];
    __syncthreads();
  }
}

__global__ void gather_xyz_kernel(const float* __restrict__ xyz, const int* __restrict__ idx,
                                  float* __restrict__ out, int M, int N, int total) {
  const int g = blockIdx.x * blockDim.x + threadIdx.x; // b*M+m
  if (g >= total) return;
  const int b = g / M;
  const int s = idx[g];
  out[g * 3 + 0] = xyz[((size_t)b * N + s) * 3 + 0];
  out[g * 3 + 1] = xyz[((size_t)b * N + s) * 3 + 1];
  out[g * 3 + 2] = xyz[((size_t)b * N + s) * 3 + 2];
}

// Ball query: thread per center; first K indices (ascending) within radius, pad with first.
__global__ void ball_query_kernel(const float* __restrict__ xyz, const float* __restrict__ cent,
                                  int N, int M, int K, float r2,
                                  int* __restrict__ out, int total) {
  const int g = blockIdx.x * blockDim.x + threadIdx.x; // b*M+m
  if (g >= total) return;
  const int b = g / M;
  const float* X = xyz + (size_t)b * N * 3;
  const float cx = cent[g * 3 + 0], cy = cent[g * 3 + 1], cz = cent[g * 3 + 2];
  int* o = out + (size_t)g * K;
  int cnt = 0;
  for (int i = 0; i < N && cnt < K; ++i) {
    const float dx = X[i * 3 + 0] - cx, dy = X[i * 3 + 1] - cy, dz = X[i * 3 + 2] - cz;
    if (dx * dx + dy * dy + dz * dz <= r2) o[cnt++] = i;
  }
  const int f = (cnt > 0) ? o[0] : 0;
  for (; cnt < K; ++cnt) o[cnt] = f;
}

// ---------------------------------------------------------------------------
// Fused set-abstraction: one block per group (b,m). Gather K neighbors to LDS,
// run 3 WMMA MLP layers in LDS, max-pool over K, write [B,M,C3].
// ---------------------------------------------------------------------------
__global__ void sa_fused_kernel(const float* __restrict__ xyz, const float* __restrict__ pts,
                                int Cpts, const float* __restrict__ cent,
                                const int* __restrict__ bq, int N, int M, int K,
                                const float* W1, const float* G1, const float* B1, int C1,
                                const float* W2, const float* G2, const float* B2, int C2,
                                const float* W3, const float* G3, const float* B3, int C3,
                                int Cin, int CinPad, float* __restrict__ outPts) {
  extern __shared__ float smem[];
  float* X0 = smem;                 // K x CinPad
  float* H1 = X0 + K * CinPad;      // K x C1
  float* H2 = H1 + K * C1;          // K x C2
  float* H3 = H2 + K * C2;          // K x C3
  const int g = blockIdx.x;         // b*M + m
  const int b = g / M;
  const int* nb = bq + (size_t)g * K;
  const float cx = cent[g * 3 + 0], cy = cent[g * 3 + 1], cz = cent[g * 3 + 2];
  for (int e = threadIdx.x; e < K * CinPad; e += blockDim.x) {
    const int k = e / CinPad, c = e - k * CinPad;
    const int src = nb[k];
    float v = 0.f;
    if (c < 3) {
      v = xyz[((size_t)b * N + src) * 3 + c] - ((c == 0) ? cx : (c == 1) ? cy : cz);
    } else if (c < 3 + Cpts) {
      v = pts[((size_t)b * N + src) * Cpts + (c - 3)];
    }
    X0[e] = v;
  }
  __syncthreads();
  mlp_layer_lds(X0, CinPad, Cin, CinPad, W1, G1, B1, H1, C1, K >> 4);
  __syncthreads();
  mlp_layer_lds(H1, C1, C1, C1, W2, G2, B2, H2, C2, K >> 4);
  __syncthreads();
  mlp_layer_lds(H2, C2, C2, C2, W3, G3, B3, H3, C3, K >> 4);
  __syncthreads();
  for (int c = threadIdx.x; c < C3; c += blockDim.x) {
    float m = H3[c];
    for (int k = 1; k < K; ++k) m = fmaxf(m, H3[k * C3 + c]);
    outPts[(size_t)g * C3 + c] = m;
  }
}

// ---------------------------------------------------------------------------
// SA3 (group_all): one block per batch; 128 rows in chunks of 32, layer-3 max
// fused into WMMA epilogue (shfl_xor(16) merges the two row-halves per lane).
// ---------------------------------------------------------------------------
__global__ void sa3_kernel(const float* __restrict__ xyz2, const float* __restrict__ pts2,
                           const float* W1, const float* G1, const float* B1,
                           const float* W2, const float* G2, const float* B2,
                           const float* W3, const float* G3, const float* B3,
                           float* __restrict__ out) {
  const int NP = 128, CH = 32;
  const int Cin = 259, CinPad = 260, C1 = 256, C2 = 512, C3 = 1024;
  extern __shared__ float smem[];
  float* X0 = smem;                  // CH x CinPad
  float* H1 = X0 + CH * CinPad;      // CH x C1
  float* H2 = H1 + CH * C1;          // CH x C2
  float* pm = H2 + CH * C2;          // C3
  const int b = blockIdx.x;
  for (int c = threadIdx.x; c < C3; c += blockDim.x) pm[c] = 0.f;
  __syncthreads();
  for (int ch = 0; ch < NP / CH; ++ch) {
    for (int e = threadIdx.x; e < CH * CinPad; e += blockDim.x) {
      const int r = e / CinPad, c = e - r * CinPad;
      const int p = ch * CH + r;
      float v = 0.f;
      if (c < 3) v = xyz2[((size_t)b * NP + p) * 3 + c];
      else if (c < 3 + 256) v = pts2[((size_t)b * NP + p) * 256 + (c - 3)];
      X0[e] = v;
    }
    __syncthreads();
    mlp_layer_lds(X0, CinPad, Cin, CinPad, W1, G1, B1, H1, C1, CH >> 4);
    __syncthreads();
    mlp_layer_lds(H1, C1, C1, C1, W2, G2, B2, H2, C2, CH >> 4);
    __syncthreads();
    {
      const int lane = threadIdx.x & 31, wave = threadIdx.x >> 5, nw = blockDim.x >> 5;
      const int ksel = (lane >> 4) << 1;
      const int ntC = C3 >> 4;
      for (int ct = wave; ct < ntC; ct += nw) {     // wave owns whole columns
        const int ncol = (ct << 4) + (lane & 15);
        const float s = G3[ncol] * INV_STD, bb = B3[ncol];
        float cmax = 0.f;
        for (int rt = 0; rt < (CH >> 4); ++rt) {
          const int mrow = (rt << 4) + (lane & 15);
          v8f acc = {0.f, 0.f, 0.f, 0.f, 0.f, 0.f, 0.f, 0.f};
          for (int k0 = 0; k0 < C2; k0 += 4) {
            const int kb = k0 + ksel;
            v2f a, bf;
            a.x  = H2[mrow * C2 + kb];
            a.y  = H2[mrow * C2 + kb + 1];
            bf.x = W3[(size_t)kb * C3 + ncol];
            bf.y = W3[(size_t)(kb + 1) * C3 + ncol];
            acc = wmma4(a, bf, acc);
          }
#pragma unroll
          for (int r = 0; r < 8; ++r)
            cmax = fmaxf(cmax, fmaxf(acc[r] * s + bb, 0.f));
        }
        cmax = fmaxf(cmax, __shfl_xor(cmax, 16, 32));
        if (lane < 16) pm[ncol] = fmaxf(pm[ncol], cmax);
      }
    }
    __syncthreads();
  }
  for (int c = threadIdx.x; c < C3; c += blockDim.x) out[(size_t)b * C3 + c] = pm[c];
}

// ---------------------------------------------------------------------------
// Generic WMMA GEMM (global->global): O = act((X @ W)*g*INV_STD + b)
// rows, Cout multiples of 16; Cin arbitrary. One wave per 16x16 tile.
// ---------------------------------------------------------------------------
__global__ void gemm_bn_act_kernel(const float* __restrict__ X, const float* __restrict__ W,
                                   const float* __restrict__ G, const float* __restrict__ Bt,
                                   float* __restrict__ O, int rows, int Cin, int Cout,
                                   int relu) {
  const int ntC = Cout >> 4;
  const int ntiles = (rows >> 4) * ntC;
  const int tile = blockIdx.x * (blockDim.x >> 5) + (threadIdx.x >> 5);
  if (tile >= ntiles) return;                   // wave-uniform exit
  const int rt = tile / ntC, ct = tile - rt * ntC;
  const int lane = threadIdx.x & 31;
  const int ksel = (lane >> 4) << 1;
  const int mrow = (rt << 4) + (lane & 15);
  const int ncol = (ct << 4) + (lane & 15);
  v8f acc = {0.f, 0.f, 0.f, 0.f, 0.f, 0.f, 0.f, 0.f};
  const int CinIter = (Cin + 3) & ~3;
  for (int k0 = 0; k0 < CinIter; k0 += 4) {
    const int kb = k0 + ksel;
    v2f a, bf;
    a.x  = (kb     < Cin) ? X[(size_t)mrow * Cin + kb]           : 0.f;
    a.y  = (kb + 1 < Cin) ? X[(size_t)mrow * Cin + kb + 1]       : 0.f;
    bf.x = (kb     < Cin) ? W[(size_t)kb * Cout + ncol]          : 0.f;
    bf.y = (kb + 1 < Cin) ? W[(size_t)(kb + 1) * Cout + ncol]    : 0.f;
    acc = wmma4(a, bf, acc);
  }
  const float s = G[ncol] * INV_STD, bb = Bt[ncol];
  const int rbase = (rt << 4) + ((lane >> 4) << 3);
#pragma unroll
  for (int r = 0; r < 8; ++r) {
    float v = acc[r] * s + bb;
    if (relu) v = fmaxf(v, 0.f);
    O[(size_t)(rbase + r) * Cout + ncol] = v;
  }
}

// 3-NN search: thread per target point; keep 3 smallest d2 (first-index ties).
__global__ void knn3_kernel(const float* __restrict__ xyz1, const float* __restrict__ xyz2,
                            int N1, int N2, int* __restrict__ ni, float* __restrict__ wgt,
                            int total) {
  const int g = blockIdx.x * blockDim.x + threadIdx.x; // b*N1+n
  if (g >= total) return;
  const int b = g / N1;
  const float px = xyz1[g * 3 + 0], py = xyz1[g * 3 + 1], pz = xyz1[g * 3 + 2];
  const float* S = xyz2 + (size_t)b * N2 * 3;
  float d0 = 1e30f, d1 = 1e30f, d2 = 1e30f;
  int i0 = 0, i1 = 0, i2 = 0;
  for (int j = 0; j < N2; ++j) {
    const float dx = px - S[j * 3 + 0], dy = py - S[j * 3 + 1], dz = pz - S[j * 3 + 2];
    const float d = dx * dx + dy * dy + dz * dz;
    if (d < d0)      { d2 = d1; i2 = i1; d1 = d0; i1 = i0; d0 = d; i0 = j; }
    else if (d < d1) { d2 = d1; i2 = i1; d1 = d; i1 = j; }
    else if (d < d2) { d2 = d; i2 = j; }
  }
  const float w0 = 1.f / (sqrtf(d0 + 1e-12f) + 1e-8f);
  const float w1 = 1.f / (sqrtf(d1 + 1e-12f) + 1e-8f);
  const float w2 = 1.f / (sqrtf(d2 + 1e-12f) + 1e-8f);
  const float ws = w0 + w1 + w2;
  ni[g * 3 + 0] = i0; ni[g * 3 + 1] = i1; ni[g * 3 + 2] = i2;
  wgt[g * 3 + 0] = w0 / ws; wgt[g * 3 + 1] = w1 / ws; wgt[g * 3 + 2] = w2 / ws;
}

// out[b,n,:] = [ interp(src via 3-NN) (Csrc) | p1a (C1a) | p1b (C1b) ]
__global__ void interp_concat_kernel(const float* __restrict__ src, int Csrc,
                                     const int* __restrict__ ni, const float* __restrict__ wgt,
                                     const float* __restrict__ p1a, int C1a,
                                     const float* __restrict__ p1b, int C1b,
                                     float* __restrict__ out, int N1, int N2, int Ctot,
                                     unsigned int total) {
  const unsigned int g = blockIdx.x * blockDim.x + threadIdx.x;
  if (g >= total) return;
  const int c = g % Ctot;
  const unsigned int bn = g / Ctot;      // b*N1 + n
  const int b = bn / N1;
  float v;
  if (c < Csrc) {
    const int j0 = ni[bn * 3 + 0], j1 = ni[bn * 3 + 1], j2 = ni[bn * 3 + 2];
    v = wgt[bn * 3 + 0] * src[((size_t)b * N2 + j0) * Csrc + c] +
        wgt[bn * 3 + 1] * src[((size_t)b * N2 + j1) * Csrc + c] +
        wgt[bn * 3 + 2] * src[((size_t)b * N2 + j2) * Csrc + c];
  } else if (c - Csrc < C1a) {
    v = p1a[(size_t)bn * C1a + (c - Csrc)];
  } else {
    v = p1b[(size_t)bn * C1b + (c - Csrc - C1a)];
  }
  out[g] = v;
}

// FP1: N2==1 broadcast: out[b,n,:] = [ l3(1024) | l2_points(256) ]
__global__ void fp1_build_kernel(const float* __restrict__ l3, const float* __restrict__ l2p,
                                 float* __restrict__ out, unsigned int total) {
  const unsigned int g = blockIdx.x * blockDim.x + threadIdx.x;
  if (g >= total) return;
  const int c = g % 1280;
  const unsigned int bn = g / 1280;   // b*128 + n
  const int b = bn / 128;
  out[g] = (c < 1024) ? l3[(size_t)b * 1024 + c]
                      : l2p[(size_t)bn * 256 + (c - 1024)];
}

// Head: out[b,n] = sigmoid(h[b,n,:] . w2 + b2)
__global__ void head_kernel(const float* __restrict__ h, const float* __restrict__ w2,
                            const float* __restrict__ b2, float* __restrict__ out, int total) {
  const int g = blockIdx.x * blockDim.x + threadIdx.x;
  if (g >= total) return;
  float acc = b2[0];
  for (int c = 0; c < 128; ++c) acc += h[(size_t)g * 128 + c] * w2[c];
  out[g] = 1.f / (1.f + expf(-acc));
}

// ---------------------------------------------------------------------------
extern "C" void kernel_launch(void* const* d_in, const int* in_sizes, int n_in,
                              void* d_out, int out_size, void* d_ws, size_t ws_size,
                              hipStream_t stream) {
  (void)in_sizes; (void)n_in; (void)out_size; (void)ws_size;
  const int B = 8, N = 4096;

  // ---- input mapping (JAX pytree: top dict insertion order, nested dict sorted keys) ----
  const float* l0_xyz  = (const float*)d_in[0];
  const float* l0_pts  = (const float*)d_in[1];
  const float* bn1_g   = (const float*)d_in[2];
  const float* bn1_b   = (const float*)d_in[3];
  const float* conv1W  = (const float*)d_in[4];
  const float* conv2_b = (const float*)d_in[5];
  const float* conv2_w = (const float*)d_in[6];
  // fp1: 2 layers (W,g,b)
  const float* fp1W1 = (const float*)d_in[7],  *fp1g1 = (const float*)d_in[8],  *fp1b1 = (const float*)d_in[9];
  const float* fp1W2 = (const float*)d_in[10], *fp1g2 = (const float*)d_in[11], *fp1b2 = (const float*)d_in[12];
  // fp2: 2 layers
  const float* fp2W1 = (const float*)d_in[13], *fp2g1 = (const float*)d_in[14], *fp2b1 = (const float*)d_in[15];
  const float* fp2W2 = (const float*)d_in[16], *fp2g2 = (const float*)d_in[17], *fp2b2 = (const float*)d_in[18];
  // fp3: 3 layers
  const float* fp3W1 = (const float*)d_in[19], *fp3g1 = (const float*)d_in[20], *fp3b1 = (const float*)d_in[21];
  const float* fp3W2 = (const float*)d_in[22], *fp3g2 = (const float*)d_in[23], *fp3b2 = (const float*)d_in[24];
  const float* fp3W3 = (const float*)d_in[25], *fp3g3 = (const float*)d_in[26], *fp3b3 = (const float*)d_in[27];
  // sa1: 3 layers
  const float* sa1W1 = (const float*)d_in[28], *sa1g1 = (const float*)d_in[29], *sa1b1 = (const float*)d_in[30];
  const float* sa1W2 = (const float*)d_in[31], *sa1g2 = (const float*)d_in[32], *sa1b2 = (const float*)d_in[33];
  const float* sa1W3 = (const float*)d_in[34], *sa1g3 = (const float*)d_in[35], *sa1b3 = (const float*)d_in[36];
  // sa2
  const float* sa2W1 = (const float*)d_in[37], *sa2g1 = (const float*)d_in[38], *sa2b1 = (const float*)d_in[39];
  const float* sa2W2 = (const float*)d_in[40], *sa2g2 = (const float*)d_in[41], *sa2b2 = (const float*)d_in[42];
  const float* sa2W3 = (const float*)d_in[43], *sa2g3 = (const float*)d_in[44], *sa2b3 = (const float*)d_in[45];
  // sa3
  const float* sa3W1 = (const float*)d_in[46], *sa3g1 = (const float*)d_in[47], *sa3b1 = (const float*)d_in[48];
  const float* sa3W2 = (const float*)d_in[49], *sa3g2 = (const float*)d_in[50], *sa3b2 = (const float*)d_in[51];
  const float* sa3W3 = (const float*)d_in[52], *sa3g3 = (const float*)d_in[53], *sa3b3 = (const float*)d_in[54];

  // ---- workspace layout (float units) ----
  float* ws = (float*)d_ws;
  size_t off = 0;
  float* FPS_DIST = ws + off; off += (size_t)B * N;         // 32768
  int*   L1_IDX   = (int*)(ws + off); off += (size_t)B * 512;
  float* L1_XYZ   = ws + off; off += (size_t)B * 512 * 3;
  int*   BQ1      = (int*)(ws + off); off += (size_t)B * 512 * 64;
  float* L1_PTS   = ws + off; off += (size_t)B * 512 * 128;
  int*   L2_IDX   = (int*)(ws + off); off += (size_t)B * 128;
  float* L2_XYZ   = ws + off; off += (size_t)B * 128 * 3;
  int*   BQ2      = (int*)(ws + off); off += (size_t)B * 128 * 64;
  float* L2_PTS   = ws + off; off += (size_t)B * 128 * 256;
  float* L3_PTS   = ws + off; off += (size_t)B * 1024;
  float* FP1_IN   = ws + off; off += (size_t)B * 128 * 1280;
  float* FP1_H    = ws + off; off += (size_t)B * 128 * 256;
  float* L2P_NEW  = ws + off; off += (size_t)B * 128 * 256;
  int*   KNN2_I   = (int*)(ws + off); off += (size_t)B * 512 * 3;
  float* KNN2_W   = ws + off; off += (size_t)B * 512 * 3;
  float* FP2_IN   = ws + off; off += (size_t)B * 512 * 384;
  float* FP2_H    = ws + off; off += (size_t)B * 512 * 256;
  float* L1P_NEW  = ws + off; off += (size_t)B * 512 * 128;
  int*   KNN3_I   = (int*)(ws + off); off += (size_t)B * N * 3;
  float* KNN3_W   = ws + off; off += (size_t)B * N * 3;
  float* FP3_IN   = ws + off; off += (size_t)B * N * 134;
  float* FP3_H1   = ws + off; off += (size_t)B * N * 128;
  float* FP3_H2   = ws + off; off += (size_t)B * N * 128;
  float* L0F      = ws + off; off += (size_t)B * N * 128;
  float* CONV1_O  = FP3_H1;  // reuse (dead by then)

  // ---- dynamic LDS sizes ----
  const int smem_sa1 = (64 * 8 + 64 * 64 + 64 * 64 + 64 * 128) * 4;        //  67,584 B
  const int smem_sa2 = (64 * 132 + 64 * 128 + 64 * 128 + 64 * 256) * 4;    // 164,864 B
  const int smem_sa3 = (32 * 260 + 32 * 256 + 32 * 512 + 1024) * 4;        // 135,680 B
  hipFuncSetAttribute((const void*)sa_fused_kernel,
                      hipFuncAttributeMaxDynamicSharedMemorySize, smem_sa2);
  hipFuncSetAttribute((const void*)sa3_kernel,
                      hipFuncAttributeMaxDynamicSharedMemorySize, smem_sa3);

  const int TB = 256;
  #define GRID1D(total) (((total) + TB - 1) / TB)

  // ======== SA1: N=4096 -> M=512, r=0.2, K=64, MLP [6,64,64,128] ========
  fps_kernel<<<B, 256, 0, stream>>>(l0_xyz, N, 512, L1_IDX, FPS_DIST);
  gather_xyz_kernel<<<GRID1D(B * 512), TB, 0, stream>>>(l0_xyz, L1_IDX, L1_XYZ, 512, N, B * 512);
  ball_query_kernel<<<GRID1D(B * 512), TB, 0, stream>>>(l0_xyz, L1_XYZ, N, 512, 64, 0.04f, BQ1, B * 512);
  sa_fused_kernel<<<B * 512, 128, smem_sa1, stream>>>(
      l0_xyz, l0_pts, 3, L1_XYZ, BQ1, N, 512, 64,
      sa1W1, sa1g1, sa1b1, 64, sa1W2, sa1g2, sa1b2, 64, sa1W3, sa1g3, sa1b3, 128,
      6, 8, L1_PTS);

  // ======== SA2: 512 -> 128, r=0.4, K=64, MLP [131,128,128,256] ========
  fps_kernel<<<B, 256, 0, stream>>>(L1_XYZ, 512, 128, L2_IDX, FPS_DIST);
  gather_xyz_kernel<<<GRID1D(B * 128), TB, 0, stream>>>(L1_XYZ, L2_IDX, L2_XYZ, 128, 512, B * 128);
  ball_query_kernel<<<GRID1D(B * 128), TB, 0, stream>>>(L1_XYZ, L2_XYZ, 512, 128, 64, 0.16f, BQ2, B * 128);
  sa_fused_kernel<<<B * 128, 128, smem_sa2, stream>>>(
      L1_XYZ, L1_PTS, 128, L2_XYZ, BQ2, 512, 128, 64,
      sa2W1, sa2g1, sa2b1, 128, sa2W2, sa2g2, sa2b2, 128, sa2W3, sa2g3, sa2b3, 256,
      131, 132, L2_PTS);

  // ======== SA3: group_all, MLP [259,256,512,1024] ========
  sa3_kernel<<<B, 128, smem_sa3, stream>>>(
      L2_XYZ, L2_PTS, sa3W1, sa3g1, sa3b1, sa3W2, sa3g2, sa3b2, sa3W3, sa3g3, sa3b3, L3_PTS);

  // ======== FP1: broadcast l3 -> l2, MLP [1280,256,256] ========
  {
    unsigned int tot = (unsigned int)B * 128 * 1280;
    fp1_build_kernel<<<GRID1D(tot), TB, 0, stream>>>(L3_PTS, L2_PTS, FP1_IN, tot);
    int tiles1 = (B * 128 / 16) * (256 / 16);
    gemm_bn_act_kernel<<<(tiles1 + 3) / 4, 128, 0, stream>>>(FP1_IN, fp1W1, fp1g1, fp1b1, FP1_H, B * 128, 1280, 256, 1);
    gemm_bn_act_kernel<<<(tiles1 + 3) / 4, 128, 0, stream>>>(FP1_H, fp1W2, fp1g2, fp1b2, L2P_NEW, B * 128, 256, 256, 1);
  }

  // ======== FP2: l2 -> l1 (3-NN), MLP [384,256,128] ========
  {
    knn3_kernel<<<GRID1D(B * 512), TB, 0, stream>>>(L1_XYZ, L2_XYZ, 512, 128, KNN2_I, KNN2_W, B * 512);
    unsigned int tot = (unsigned int)B * 512 * 384;
    interp_concat_kernel<<<GRID1D(tot), TB, 0, stream>>>(
        L2P_NEW, 256, KNN2_I, KNN2_W, L1_PTS, 128, (const float*)nullptr, 0,
        FP2_IN, 512, 128, 384, tot);
    int tilesA = (B * 512 / 16) * (256 / 16);
    int tilesB = (B * 512 / 16) * (128 / 16);
    gemm_bn_act_kernel<<<(tilesA + 3) / 4, 128, 0, stream>>>(FP2_IN, fp2W1, fp2g1, fp2b1, FP2_H, B * 512, 384, 256, 1);
    gemm_bn_act_kernel<<<(tilesB + 3) / 4, 128, 0, stream>>>(FP2_H, fp2W2, fp2g2, fp2b2, L1P_NEW, B * 512, 256, 128, 1);
  }

  // ======== FP3: l1 -> l0 (3-NN), points1=[l0_points|l0_xyz], MLP [134,128,128,128] ========
  {
    knn3_kernel<<<GRID1D(B * N), TB, 0, stream>>>(l0_xyz, L1_XYZ, N, 512, KNN3_I, KNN3_W, B * N);
    unsigned int tot = (unsigned int)B * N * 134;
    interp_concat_kernel<<<GRID1D(tot), TB, 0, stream>>>(
        L1P_NEW, 128, KNN3_I, KNN3_W, l0_pts, 3, l0_xyz, 3,
        FP3_IN, N, 512, 134, tot);
    int tiles = (B * N / 16) * (128 / 16);
    gemm_bn_act_kernel<<<(tiles + 3) / 4, 128, 0, stream>>>(FP3_IN, fp3W1, fp3g1, fp3b1, FP3_H1, B * N, 134, 128, 1);
    gemm_bn_act_kernel<<<(tiles + 3) / 4, 128, 0, stream>>>(FP3_H1, fp3W2, fp3g2, fp3b2, FP3_H2, B * N, 128, 128, 1);
    gemm_bn_act_kernel<<<(tiles + 3) / 4, 128, 0, stream>>>(FP3_H2, fp3W3, fp3g3, fp3b3, L0F,    B * N, 128, 128, 1);
    // conv1 + bn1 + relu
    gemm_bn_act_kernel<<<(tiles + 3) / 4, 128, 0, stream>>>(L0F, conv1W, bn1_g, bn1_b, CONV1_O, B * N, 128, 128, 1);
  }

  // ======== head: sigmoid(conv2) -> d_out [B,1,N] ========
  head_kernel<<<GRID1D(B * N), TB, 0, stream>>>(CONV1_O, conv2_w, conv2_b, (float*)d_out, B * N);
  #undef GRID1D
}